// DecoderBlock_84739704750213
// MI455X (gfx1250) — compile-verified
//
#include <hip/hip_runtime.h>
#include <hip/hip_bf16.h>
#include <math.h>

typedef __attribute__((ext_vector_type(2))) float v2f;
typedef __attribute__((ext_vector_type(4))) float v4f;
typedef __attribute__((ext_vector_type(8))) float v8f;

#define B_SZ   256
#define NFEAT  8192           // C_OUT * F_OUT = 32*256
#define BLK_K  1024           // per channel-block K = 4*256
#define BLK_N  1024           // per channel-block N = 4*256
#define KC     32             // K chunk staged in LDS
#define TILE_M 64
#define TILE_N 128
#define LDA    36             // padded LDS row stride (floats), conflict-free col reads
#define LDB    36
#define W_OSTRIDE 2097152     // 32*256*256
#define W_CSTRIDE 65536       // 256*256

// ---------------------------------------------------------------------------
// Kernel 1: block-diagonal masked GEMM + bias, fp32 WMMA (V_WMMA_F32_16X16X4_F32)
// y[b, g*1024 + n] = sum_k A[b][k] * Bmat[n][k] + bias,  per block g
// ---------------------------------------------------------------------------
__global__ __launch_bounds__(256) void dec_gemm_kernel(
    const float* __restrict__ x, const float* __restrict__ W,
    const float* __restrict__ bias, float* __restrict__ y)
{
    __shared__ float ldsA[TILE_M * LDA];   // [row 0..63][k 0..31], stride 36
    __shared__ float ldsB[TILE_N * LDB];   // [n   0..127][k 0..31], stride 36

    const int tid  = threadIdx.x;
    const int lane = tid & 31;
    const int wave = tid >> 5;          // 0..7
    const int mw   = wave & 3;          // 4 M-subtiles of 16 rows
    const int nw   = wave >> 2;         // 2 N-subtiles of 64 cols

    const int bid   = blockIdx.x;       // 256 workgroups total
    const int g     = bid >> 5;         // channel block 0..7
    const int mTile = (bid >> 3) & 3;   // 4 tiles of 64 batch rows
    const int nTile = bid & 7;          // 8 tiles of 128 cols

    const int m0 = mTile * TILE_M;      // batch-row base
    const int n0 = nTile * TILE_N;      // block-local col base

    // cooperative-load coordinates (float4 granularity)
    const int ldk = (tid & 7) * 4;      // 0,4,...,28 within K chunk
    const int ldr = tid >> 3;           // 0..31

    // A[b][k] = x[b*8192 + g*1024 + k]
    const float* Abase = x + (size_t)m0 * NFEAT + (size_t)g * BLK_K;

    v8f acc[4];
#pragma unroll
    for (int a = 0; a < 4; ++a) acc[a] = (v8f)0.f;

    // WMMA f32 16x16x4 lane mapping (ISA 7.12.2):
    //   A: lane -> M = lane&15 ; VGPR v holds K = 2*(lane>>4) + v
    //   B: lane -> N = lane&15 ; VGPR v holds K = 2*(lane>>4) + v
    const int mloc = lane & 15;
    const int kloc = (lane >> 4) * 2;

    for (int k0 = 0; k0 < BLK_K; k0 += KC) {
        __syncthreads();
        // ---- stage A tile: 64 rows x 32 k (2 float4 per thread) ----
#pragma unroll
        for (int r = 0; r < 2; ++r) {
            const int row = r * 32 + ldr;
            v4f a4 = *(const v4f*)(Abase + (size_t)row * NFEAT + k0 + ldk);
            *(v4f*)&ldsA[row * LDA + ldk] = a4;   // row*36*4B is 16B aligned
        }
        // ---- stage B tile: 128 n x 32 k, stored [n][k] (4 float4 per thread) ----
        // W[o,c,f,i]: for block-local col nn: o_local=nn>>8, f=nn&255;
        // K chunk of 32 stays inside one c_local (=k0>>8), i run is contiguous.
        const size_t cbase = (size_t)(g * 4 + (k0 >> 8)) * W_CSTRIDE + (k0 & 255);
#pragma unroll
        for (int r = 0; r < 4; ++r) {
            const int n  = r * 32 + ldr;
            const int nn = n0 + n;
            const float* wn = W + (size_t)(g * 4 + (nn >> 8)) * W_OSTRIDE
                                + cbase + (size_t)(nn & 255) * 256;
            v4f b4 = *(const v4f*)(wn + ldk);
            *(v4f*)&ldsB[n * LDB + ldk] = b4;
        }
        __syncthreads();

        // speculative prefetch of next K chunk (global_prefetch_b8)
        if (k0 + KC < BLK_K) {
            __builtin_prefetch(Abase + (size_t)ldr * NFEAT + k0 + KC + ldk, 0, 0);
            const size_t cb2 = (size_t)(g * 4 + ((k0 + KC) >> 8)) * W_CSTRIDE + ((k0 + KC) & 255);
            const int nn2 = n0 + ldr;
            __builtin_prefetch(W + (size_t)(g * 4 + (nn2 >> 8)) * W_OSTRIDE
                                 + cb2 + (size_t)(nn2 & 255) * 256 + ldk, 0, 0);
        }

        // ---- compute: 8 K-steps x 4 accumulators of v_wmma_f32_16x16x4_f32 ----
#pragma unroll
        for (int ks = 0; ks < KC; ks += 4) {
            v2f a2 = *(const v2f*)&ldsA[(mw * 16 + mloc) * LDA + ks + kloc];
#pragma unroll
            for (int a = 0; a < 4; ++a) {
                v2f b2 = *(const v2f*)&ldsB[(nw * 64 + a * 16 + mloc) * LDB + ks + kloc];
                acc[a] = __builtin_amdgcn_wmma_f32_16x16x4_f32(
                    /*neg_a=*/false, a2, /*neg_b=*/false, b2,
                    /*c_mod=*/(short)0, acc[a], /*reuse_a=*/false, /*reuse_b=*/false);
            }
        }
    }

    // ---- epilogue: add bias, scatter-store (C/D layout: VGPR r -> M = r + 8*(lane>>4)) ----
    const int row_hi = (lane >> 4) * 8;
    const int coln   = lane & 15;
#pragma unroll
    for (int a = 0; a < 4; ++a) {
        const int jn = n0 + nw * 64 + a * 16 + coln;  // block-local feature col
        const int j  = g * BLK_N + jn;                // global feature index (o*256+f)
        const float bsv = bias[j];
#pragma unroll
        for (int r = 0; r < 8; ++r) {
            const int brow = m0 + mw * 16 + row_hi + r;
            y[(size_t)brow * NFEAT + j] = acc[a][r] + bsv;
        }
    }
}

// ---------------------------------------------------------------------------
// Kernel 2: per-feature batch statistics (biased), lane = feature -> coalesced
// ---------------------------------------------------------------------------
__global__ __launch_bounds__(256) void dec_bnstats_kernel(
    const float* __restrict__ y, float* __restrict__ meanArr, float* __restrict__ invArr)
{
    const int j = blockIdx.x * 256 + threadIdx.x;
    float s = 0.f, s2 = 0.f;
    for (int b = 0; b < B_SZ; ++b) {
        const float v = y[(size_t)b * NFEAT + j];
        s += v; s2 += v * v;
    }
    const float mean = s * (1.f / B_SZ);
    const float var  = fmaxf(s2 * (1.f / B_SZ) - mean * mean, 0.f);
    meanArr[j] = mean;
    invArr[j]  = rsqrtf(var + 1e-5f);
}

// ---------------------------------------------------------------------------
// Kernel 3: normalize + affine + Swish, float4, in place (y fully rewritten by K1)
// ---------------------------------------------------------------------------
__global__ __launch_bounds__(256) void dec_bnswish_kernel(
    float* __restrict__ y, const float* __restrict__ meanArr,
    const float* __restrict__ invArr, const float* __restrict__ gamma,
    const float* __restrict__ beta)
{
    const int idx = blockIdx.x * 256 + threadIdx.x;
    const int e   = idx * 4;
    const int j   = e & (NFEAT - 1);      // float4 never crosses a feature-row boundary
    v4f v  = *(v4f*)(y + e);
    v4f mn = *(const v4f*)(meanArr + j);
    v4f iv = *(const v4f*)(invArr + j);
    v4f gm = *(const v4f*)(gamma + j);
    v4f bt = *(const v4f*)(beta + j);
#pragma unroll
    for (int t = 0; t < 4; ++t) {
        const float z = (v[t] - mn[t]) * iv[t] * gm[t] + bt[t];
        v[t] = z / (1.f + __expf(-z));    // z * sigmoid(z)
    }
    *(v4f*)(y + e) = v;
}

// ---------------------------------------------------------------------------
extern "C" void kernel_launch(void* const* d_in, const int* in_sizes, int n_in,
                              void* d_out, int out_size, void* d_ws, size_t ws_size,
                              hipStream_t stream) {
    const float* x     = (const float*)d_in[0];
    const float* W     = (const float*)d_in[1];
    const float* bias  = (const float*)d_in[2];
    const float* gamma = (const float*)d_in[3];
    const float* beta  = (const float*)d_in[4];
    // d_in[5] = mask: implied by the block-diagonal GEMM decomposition, unused.
    float* y = (float*)d_out;
    float* meanArr = (float*)d_ws;
    float* invArr  = meanArr + NFEAT;

    dec_gemm_kernel<<<256, 256, 0, stream>>>(x, W, bias, y);
    dec_bnstats_kernel<<<NFEAT / 256, 256, 0, stream>>>(y, meanArr, invArr);
    dec_bnswish_kernel<<<(B_SZ * NFEAT) / (256 * 4), 256, 0, stream>>>(y, meanArr, invArr, gamma, beta);
    (void)in_sizes; (void)n_in; (void)out_size; (void)ws_size;
}